// PerTokenSparseMoE_49091476193895
// MI455X (gfx1250) — compile-verified
//
#include <hip/hip_runtime.h>
#include <stdint.h>

// ---------------------------------------------------------------------------
// PerTokenSparseMoE fused kernel for gfx1250 (MI455X).
//  * fp32 WMMA (V_WMMA_F32_16X16X4_F32) -> bit-compatible with fp32 reference
//  * weights staged TRANSPOSED into LDS via global_load_async_to_lds_b32
//    (per-lane scatter == free transpose), double-buffered, synced with
//    s_wait_asynccnt; X tile staged via global_load_async_to_lds_b128.
//  * gate folded into h before GEMM2; single pass over y accumulators.
// ---------------------------------------------------------------------------

#define BQ 2048
#define TQ 16
#define DQ 128
#define EQ 4
#define HQ 512

typedef float v2f __attribute__((ext_vector_type(2)));
typedef float v8f __attribute__((ext_vector_type(8)));

// Strides padded so stride % 64 == 4 -> 16 parallel column reads hit distinct
// LDS banks; stride*4 bytes is a multiple of 16 for b64/b128 alignment.
#define XS_STRIDE  132   // X tile rows:   [128][132]
#define HS_STRIDE  68    // h chunk rows:  [128][68]
#define W1T_STRIDE 132   // W1 chunk ^T:   [64 n][132 k]  (k = 0..127)
#define W2T_STRIDE 68    // W2 chunk ^T:   [128 n][68 k]  (k = 0..63)

#define XS_SIZE   (128 * XS_STRIDE)   // 16896 floats
#define HS_SIZE   (128 * HS_STRIDE)   // 8704
#define WB_SIZE   8704                // >= max(64*132=8448, 128*68=8704)
#define GWS_SIZE  (DQ * EQ)           // 512
#define GATES_SIZE (128 * EQ)         // 512
#define SMEM_FLOATS (XS_SIZE + HS_SIZE + 2 * WB_SIZE + GWS_SIZE + GATES_SIZE)

__device__ __forceinline__ v8f wmma_f32(v2f a, v2f b, v8f c) {
  return __builtin_amdgcn_wmma_f32_16x16x4_f32(
      false, a, false, b, (short)0, c, false, false);
}

__device__ __forceinline__ float gelu_exact(float h) {
  return 0.5f * h * (1.0f + erff(h * 0.70710678118654752f));
}

// ---- CDNA5 async global->LDS copies (ASYNCcnt-tracked) ----
__device__ __forceinline__ void async_ld_b32(uint32_t lds_byte, uint32_t voff,
                                             uint64_t sbase) {
  asm volatile("global_load_async_to_lds_b32 %0, %1, %2"
               :
               : "v"(lds_byte), "v"(voff), "s"(sbase)
               : "memory");
}
__device__ __forceinline__ void async_ld_b128(uint32_t lds_byte, uint32_t voff,
                                              uint64_t sbase) {
  asm volatile("global_load_async_to_lds_b128 %0, %1, %2"
               :
               : "v"(lds_byte), "v"(voff), "s"(sbase)
               : "memory");
}
__device__ __forceinline__ void wait_async_le32() {
  asm volatile("s_wait_asynccnt 0x20" ::: "memory");
}

// Stage W1[t,e][:, hc:hc+64] transposed: Wt[n][k], n<64, k<128.
__device__ __forceinline__ void prefetch_w1t(const float* W1e_hc, int tid,
                                             uint32_t wb_byte) {
  const int n  = tid & 63;
  const int k0 = tid >> 6;  // 0..3, covers k via k0 + 4*it
  uint32_t lds  = wb_byte + (uint32_t)((n * W1T_STRIDE + k0) * 4);
  uint32_t voff = (uint32_t)((k0 * HQ + n) * 4);
  const uint64_t base = (uint64_t)(uintptr_t)W1e_hc;
#pragma unroll
  for (int it = 0; it < 32; ++it) {
    async_ld_b32(lds, voff, base);
    lds  += 4 * 4;           // k += 4
    voff += 4 * HQ * 4;
  }
}

// Stage W2[t,e][hc:hc+64, :] transposed: Wt[n][k], n<128, k<64.
__device__ __forceinline__ void prefetch_w2t(const float* W2e_hc, int tid,
                                             uint32_t wb_byte) {
  const int n  = tid & 127;
  const int k0 = tid >> 7;  // 0..1, covers k via k0 + 2*it
  uint32_t lds  = wb_byte + (uint32_t)((n * W2T_STRIDE + k0) * 4);
  uint32_t voff = (uint32_t)((k0 * DQ + n) * 4);
  const uint64_t base = (uint64_t)(uintptr_t)W2e_hc;
#pragma unroll
  for (int it = 0; it < 32; ++it) {
    async_ld_b32(lds, voff, base);
    lds  += 2 * 4;           // k += 2
    voff += 2 * DQ * 4;
  }
}

__global__ __launch_bounds__(256) void moe_fused_kernel(
    const float* __restrict__ x,   // (B,T,D)
    const float* __restrict__ W1,  // (T,E,D,H)
    const float* __restrict__ b1,  // (T,E,H)
    const float* __restrict__ W2,  // (T,E,H,D)
    const float* __restrict__ b2,  // (T,E,D)
    const float* __restrict__ gw,  // (T,D,E)
    const float* __restrict__ gb,  // (T,E)
    float* __restrict__ out)       // (B,T,D) + 1 (l1 loss)
{
  extern __shared__ float smem[];
  float* Xs  = smem;                 // [128][XS_STRIDE]
  float* Hs  = Xs + XS_SIZE;         // [128][HS_STRIDE]
  float* Wb0 = Hs + HS_SIZE;         // W1 chunk transposed
  float* Wb1 = Wb0 + WB_SIZE;        // W2 chunk transposed
  float* Gw  = Wb1 + WB_SIZE;        // gate weights [D][E]
  float* Gs  = Gw + GWS_SIZE;        // gate values  [128][E]

  const int tid = threadIdx.x;
  const int bid = blockIdx.x;
  const int t   = bid & 15;
  const int r0  = (bid >> 4) * 128;

  const uint32_t xsB  = (uint32_t)(uintptr_t)Xs;
  const uint32_t wb0B = (uint32_t)(uintptr_t)Wb0;
  const uint32_t wb1B = (uint32_t)(uintptr_t)Wb1;

  if (bid == 0 && tid == 0) out[BQ * TQ * DQ] = 0.0f;  // l1_loss == 0

  // ---- async-stage X tile (128x128, b128 per lane), then W1 chunk 0 ----
  {
    const int c = (tid & 31) << 2;          // constant per thread
    const int row0 = tid >> 5;              // + 8*it
    uint32_t lds  = xsB + (uint32_t)((row0 * XS_STRIDE + c) * 4);
    uint32_t voff = (uint32_t)((row0 * (TQ * DQ) + c) * 4);
    const uint64_t base =
        (uint64_t)(uintptr_t)(x + ((size_t)r0 * TQ + t) * DQ);
#pragma unroll
    for (int it = 0; it < 16; ++it) {
      async_ld_b128(lds, voff, base);
      lds  += 8 * XS_STRIDE * 4;
      voff += 8 * (TQ * DQ) * 4;
    }
  }
  prefetch_w1t(W1 + (size_t)t * EQ * DQ * HQ, tid, wb0B);  // e=0, hc=0

  if (tid < 128) {  // gate weights (tiny, regular load)
    float4 v = *(const float4*)(gw + ((size_t)t * DQ + tid) * EQ);
    *(float4*)(Gw + tid * EQ) = v;
  }
  wait_async_le32();   // X (issued first, 16 oldest) complete
  __syncthreads();

  // ---- gate: Gs[i][e] = relu(dot(Xs[i,:], Gw[:,e]) + gb[t,e]) ----
#pragma unroll
  for (int q0 = 0; q0 < 2; ++q0) {
    int q = tid + q0 * 256;
    int i = q >> 2;
    int e = q & 3;
    float s = gb[t * EQ + e];
#pragma unroll 8
    for (int d = 0; d < DQ; ++d)
      s = fmaf(Xs[i * XS_STRIDE + d], Gw[d * EQ + e], s);
    Gs[i * EQ + e] = fmaxf(s, 0.0f);
  }
  __syncthreads();

  // ---- wave tiling ----
  const int w    = tid >> 5;
  const int lane = tid & 31;
  const int ln   = lane & 15;
  const int hi   = lane >> 4;
  const int wr   = (w >> 1) * 32;   // row offset (both GEMMs)
  const int wc1  = (w & 1) * 32;    // GEMM1 col offset in 64
  const int wc2  = (w & 1) * 64;    // GEMM2 col offset in 128

  const v8f zero8 = {0.f, 0.f, 0.f, 0.f, 0.f, 0.f, 0.f, 0.f};
  v8f yacc[2][4];
#pragma unroll
  for (int rt = 0; rt < 2; ++rt)
#pragma unroll
    for (int ct = 0; ct < 4; ++ct) yacc[rt][ct] = zero8;

  // Operand bases (inner offsets are compile-time -> DS immediates).
  const float* aBase = Xs + (wr + ln) * XS_STRIDE + 2 * hi;         // G1 A
  const float* bT1   = Wb0 + (wc1 + ln) * W1T_STRIDE + 2 * hi;      // G1 B^T
  const float* a2Base = Hs + (wr + ln) * HS_STRIDE + 2 * hi;        // G2 A
  const float* bT2   = Wb1 + (wc2 + ln) * W2T_STRIDE + 2 * hi;      // G2 B^T
  float* hsBase = Hs + (wr + 8 * hi) * HS_STRIDE + wc1 + ln;        // h store

  for (int e = 0; e < EQ; ++e) {
    const float* W1e = W1 + ((size_t)t * EQ + e) * DQ * HQ;
    const float* W2e = W2 + ((size_t)t * EQ + e) * HQ * DQ;
    const float* b1e = b1 + ((size_t)t * EQ + e) * HQ;

    for (int hcb = 0; hcb < 8; ++hcb) {
      const int hc = hcb * 64;

      // overlap: fetch W2(e,hc) while GEMM1 runs
      prefetch_w2t(W2e + (size_t)hc * DQ, tid, wb1B);
      wait_async_le32();  // W1(e,hc) (older 32) complete
      __syncthreads();

      // -- GEMM1: h(128x64) = X @ W1chunk, 4 tiles/wave --
      v8f hacc[2][2];
      hacc[0][0] = zero8; hacc[0][1] = zero8;
      hacc[1][0] = zero8; hacc[1][1] = zero8;
#pragma unroll
      for (int kk = 0; kk < 32; ++kk) {
        v2f a0 = *(const v2f*)(aBase + kk * 4);
        v2f a1 = *(const v2f*)(aBase + 16 * XS_STRIDE + kk * 4);
        v2f b0 = *(const v2f*)(bT1 + kk * 4);
        v2f b1v = *(const v2f*)(bT1 + 16 * W1T_STRIDE + kk * 4);
        hacc[0][0] = wmma_f32(a0, b0, hacc[0][0]);
        hacc[0][1] = wmma_f32(a0, b1v, hacc[0][1]);
        hacc[1][0] = wmma_f32(a1, b0, hacc[1][0]);
        hacc[1][1] = wmma_f32(a1, b1v, hacc[1][1]);
      }

      // -- epilogue: +b1, exact gelu, fold gate; store h chunk to LDS --
      {
        float bias0 = b1e[hc + wc1 + ln];
        float bias1 = b1e[hc + wc1 + 16 + ln];
#pragma unroll
        for (int rt = 0; rt < 2; ++rt) {
#pragma unroll
          for (int v = 0; v < 8; ++v) {
            float g  = Gs[(wr + rt * 16 + 8 * hi + v) * EQ + e];
            float h0 = gelu_exact(hacc[rt][0][v] + bias0) * g;
            float h1 = gelu_exact(hacc[rt][1][v] + bias1) * g;
            hsBase[(rt * 16 + v) * HS_STRIDE]      = h0;
            hsBase[(rt * 16 + v) * HS_STRIDE + 16] = h1;
          }
        }
      }
      __syncthreads();  // buf0 free; Hs visible to all waves

      // overlap: fetch next W1 chunk while GEMM2 runs (wrap at the end)
      {
        int ne = e, nh = hcb + 1;
        if (nh == 8) { nh = 0; if (++ne == EQ) ne = 0; }
        const float* W1n = W1 + ((size_t)t * EQ + ne) * DQ * HQ + nh * 64;
        prefetch_w1t(W1n, tid, wb0B);
      }
      wait_async_le32();  // W2(e,hc) complete
      __syncthreads();

      // -- GEMM2: yacc(128x128) += Hs(128x64) @ W2chunk --
#pragma unroll
      for (int kk = 0; kk < 16; ++kk) {
        v2f a0 = *(const v2f*)(a2Base + kk * 4);
        v2f a1 = *(const v2f*)(a2Base + 16 * HS_STRIDE + kk * 4);
#pragma unroll
        for (int ct = 0; ct < 4; ++ct) {
          v2f bv = *(const v2f*)(bT2 + ct * 16 * W2T_STRIDE + kk * 4);
          yacc[0][ct] = wmma_f32(a0, bv, yacc[0][ct]);
          yacc[1][ct] = wmma_f32(a1, bv, yacc[1][ct]);
        }
      }
      __syncthreads();  // buf1 + Hs free for next iteration
    }  // hcb
  }  // e

  // ---- final epilogue: y += sum_e gate[row,e]*b2[t,e,col]; store ----
  {
    float bb2[4][4];  // [ct][e]
#pragma unroll
    for (int ct = 0; ct < 4; ++ct) {
      int colg = wc2 + ct * 16 + ln;
#pragma unroll
      for (int e = 0; e < 4; ++e)
        bb2[ct][e] = b2[((size_t)t * EQ + e) * DQ + colg];
    }
    float* outBase = out + ((size_t)(r0 + wr) * TQ + t) * DQ;
#pragma unroll
    for (int rt = 0; rt < 2; ++rt) {
#pragma unroll
      for (int v = 0; v < 8; ++v) {
        int rowl = rt * 16 + 8 * hi + v;
        float4 g = *(const float4*)(Gs + (wr + rowl) * EQ);
#pragma unroll
        for (int ct = 0; ct < 4; ++ct) {
          float val = yacc[rt][ct][v];
          val = fmaf(g.x, bb2[ct][0], val);
          val = fmaf(g.y, bb2[ct][1], val);
          val = fmaf(g.z, bb2[ct][2], val);
          val = fmaf(g.w, bb2[ct][3], val);
          outBase[(size_t)rowl * (TQ * DQ) + wc2 + ct * 16 + ln] = val;
        }
      }
    }
  }
}

extern "C" void kernel_launch(void* const* d_in, const int* in_sizes, int n_in,
                              void* d_out, int out_size, void* d_ws, size_t ws_size,
                              hipStream_t stream) {
  const float* x  = (const float*)d_in[0];
  const float* W1 = (const float*)d_in[1];
  const float* b1 = (const float*)d_in[2];
  const float* W2 = (const float*)d_in[3];
  const float* b2 = (const float*)d_in[4];
  const float* gw = (const float*)d_in[5];
  const float* gb = (const float*)d_in[6];
  float* out = (float*)d_out;

  const size_t smem_bytes = (size_t)SMEM_FLOATS * sizeof(float);  // 176,128 B
  (void)hipFuncSetAttribute(reinterpret_cast<const void*>(moe_fused_kernel),
                            hipFuncAttributeMaxDynamicSharedMemorySize,
                            (int)smem_bytes);

  dim3 grid(TQ * (BQ / 128));  // 256 workgroups: (t, row-block)
  dim3 block(256);             // 8 wave32 waves
  hipLaunchKernelGGL(moe_fused_kernel, grid, block, smem_bytes, stream,
                     x, W1, b1, W2, b2, gw, gb, out);
}